// GeneralAddAttConv_36000415875690
// MI455X (gfx1250) — compile-verified
//
#include <hip/hip_runtime.h>

// GAT forward for MI455X (gfx1250, wave32).
// N=50000 nodes, E=1,600,000 edges, D=128, H=4 heads, C=32.
//
// Phase breakdown / roofline:
//  - GEMM xh = x@W: 1.64 GFLOP, done with V_WMMA_F32_16X16X4_F32 (exact fp32).
//  - Edge phase: ~0.8 GB gather + ~0.8 GB fp32 atomic scatter; xh (25.6MB) and
//    out (25.6MB) both fit in the 192MB L2, so atomics resolve on-chip.
//  - alpha[e,h] factors into s_dst[dst,h] + s_src[src,h]: O(E*C) -> O(E*H).

typedef float v2f __attribute__((ext_vector_type(2)));
typedef float v8f __attribute__((ext_vector_type(8)));

#define GAT_N 50000
#define GAT_E 1600000
#define GAT_D 128
#define GAT_H 4
#define GAT_C 32
#define GAT_SLOPE 0.2f

// ---- monotone float <-> ordered-uint mapping (for integer atomicMax) ----
__device__ __forceinline__ unsigned ord_encode(float f) {
  unsigned b = __float_as_uint(f);
  return (b & 0x80000000u) ? ~b : (b ^ 0x80000000u);
}
__device__ __forceinline__ float ord_decode(unsigned u) {
  unsigned b = (u & 0x80000000u) ? (u ^ 0x80000000u) : ~u;
  return __uint_as_float(b);
}

// ---- 0: init out / segment-max / denom --------------------------------
__global__ void gat_init(float* __restrict__ out, unsigned* __restrict__ amax_u,
                         float* __restrict__ denom, int n_out, int nh) {
  int t = blockIdx.x * blockDim.x + threadIdx.x;
  if (t < n_out) out[t] = 0.0f;
  if (t < nh) { amax_u[t] = 0u; denom[t] = 0.0f; }  // 0u == ordered -inf
}

// ---- 1: xh = x @ W via fp32 WMMA --------------------------------------
// Block: 256 threads = 8 waves. Block owns rows [16*bid, 16*bid+16).
// Wave w owns output columns [16w, 16w+16). K loop: 32 steps of K=4.
// 50000 % 16 == 0 so every block is full and EXEC is all-ones (WMMA req).
__global__ __launch_bounds__(256) void gat_gemm_wmma(
    const float* __restrict__ x, const float* __restrict__ W,
    float* __restrict__ xh) {
  __shared__ float ldsX[16 * GAT_D];  // 8 KB x-tile

  const int tid = threadIdx.x;
  const int m0 = blockIdx.x * 16;

  // stage the 16x128 x tile into LDS (coalesced, 8 floats/thread)
  for (int i = tid; i < 16 * GAT_D; i += 256) {
    ldsX[i] = x[(size_t)m0 * GAT_D + i];
  }
  __syncthreads();

  const int wave = tid >> 5;
  const int lane = tid & 31;
  const int half = lane >> 4;   // 0: lanes 0-15, 1: lanes 16-31
  const int l16  = lane & 15;
  const int col  = wave * 16 + l16;

  v8f acc = {0.f, 0.f, 0.f, 0.f, 0.f, 0.f, 0.f, 0.f};

#pragma unroll
  for (int k4 = 0; k4 < GAT_D / 4; ++k4) {
    const int kb = k4 * 4 + 2 * half;  // this half-wave's K base
    v2f a, b;
    // A 16x4 f32 layout: lane = M (l16); VGPR j holds K = 2*half + j
    a.x = ldsX[l16 * GAT_D + kb + 0];
    a.y = ldsX[l16 * GAT_D + kb + 1];
    // B 4x16 f32 layout: lane = N (l16); VGPR j holds K = 2*half + j
    b.x = W[(size_t)(kb + 0) * GAT_D + col];
    b.y = W[(size_t)(kb + 1) * GAT_D + col];
    acc = __builtin_amdgcn_wmma_f32_16x16x4_f32(
        /*neg_a=*/false, a, /*neg_b=*/false, b,
        /*c_mod=*/(short)0, acc, /*reuse_a=*/false, /*reuse_b=*/false);
  }

  // C/D 16x16 f32 layout: VGPR v -> row M = v + 8*half, col N = l16
#pragma unroll
  for (int v = 0; v < 8; ++v) {
    const int row = m0 + v + 8 * half;
    xh[(size_t)row * GAT_D + col] = acc[v];
  }
}

// ---- 2: per-node score halves  s_d[n,h]=<xh,a_i>, s_s[n,h]=<xh,a_j> ----
__global__ void gat_scores(const float* __restrict__ xh,
                           const float* __restrict__ att,
                           float* __restrict__ s_d, float* __restrict__ s_s,
                           int nh) {
  int t = blockIdx.x * blockDim.x + threadIdx.x;
  if (t >= nh) return;
  const int n = t >> 2;
  const int h = t & 3;
  const float* v  = xh + (size_t)n * GAT_D + h * GAT_C;
  const float* ai = att + h * (2 * GAT_C);       // att[0, h, 0:C]
  const float* aj = ai + GAT_C;                  // att[0, h, C:2C]
  float si = 0.f, sj = 0.f;
#pragma unroll
  for (int c = 0; c < GAT_C; ++c) {
    si += v[c] * ai[c];
    sj += v[c] * aj[c];
  }
  s_d[t] = si;
  s_s[t] = sj;
}

// ---- 3: edge pass 1 — leaky-relu + segment max (ordered-uint atomicMax)
__global__ void gat_edge_max(const int* __restrict__ srcI,
                             const int* __restrict__ dstI,
                             const float* __restrict__ s_d,
                             const float* __restrict__ s_s,
                             unsigned* __restrict__ amax_u, int ne) {
  int e = blockIdx.x * blockDim.x + threadIdx.x;
  if (e >= ne) return;
  const int s = srcI[e];
  const int d = dstI[e];
#pragma unroll
  for (int h = 0; h < GAT_H; ++h) {
    float a = s_d[d * GAT_H + h] + s_s[s * GAT_H + h];
    a = a > 0.f ? a : GAT_SLOPE * a;
    atomicMax(amax_u + d * GAT_H + h, ord_encode(a));
  }
}

// ---- 4: edge pass 2 — exp(alpha - amax) and denominator sum -----------
__global__ void gat_edge_sum(const int* __restrict__ srcI,
                             const int* __restrict__ dstI,
                             const float* __restrict__ s_d,
                             const float* __restrict__ s_s,
                             const unsigned* __restrict__ amax_u,
                             float* __restrict__ denom, int ne) {
  int e = blockIdx.x * blockDim.x + threadIdx.x;
  if (e >= ne) return;
  const int s = srcI[e];
  const int d = dstI[e];
#pragma unroll
  for (int h = 0; h < GAT_H; ++h) {
    float a = s_d[d * GAT_H + h] + s_s[s * GAT_H + h];
    a = a > 0.f ? a : GAT_SLOPE * a;
    const float ex = expf(a - ord_decode(amax_u[d * GAT_H + h]));
    atomicAdd(denom + d * GAT_H + h, ex);
  }
}

// ---- 5: edge pass 3 — scatter msg = coeff * xh[src] into out[dst] -----
// One wave32 per edge; lane owns 4 consecutive channels (one float4),
// all 4 inside one head (head = lane >> 3). L2-resident atomics.
__global__ __launch_bounds__(256) void gat_scatter(
    const float* __restrict__ xh, const int* __restrict__ srcI,
    const int* __restrict__ dstI, const float* __restrict__ s_d,
    const float* __restrict__ s_s, const unsigned* __restrict__ amax_u,
    const float* __restrict__ denom, float* __restrict__ out, int ne) {
  const int e = (int)((blockIdx.x * 256 + threadIdx.x) >> 5);
  const int lane = threadIdx.x & 31;
  if (e >= ne) return;
  const int s = srcI[e];
  const int d = dstI[e];
  const int h = lane >> 3;
  const int idx = d * GAT_H + h;

  float a = s_d[idx] + s_s[s * GAT_H + h];
  a = a > 0.f ? a : GAT_SLOPE * a;
  const float ex = expf(a - ord_decode(amax_u[idx]));
  const float coeff = ex / (denom[idx] + 1e-16f);

  const float4 xv =
      *reinterpret_cast<const float4*>(xh + (size_t)s * GAT_D + lane * 4);
  float* o = out + (size_t)d * GAT_D + lane * 4;
  atomicAdd(o + 0, xv.x * coeff);
  atomicAdd(o + 1, xv.y * coeff);
  atomicAdd(o + 2, xv.z * coeff);
  atomicAdd(o + 3, xv.w * coeff);
}

extern "C" void kernel_launch(void* const* d_in, const int* in_sizes, int n_in,
                              void* d_out, int out_size, void* d_ws,
                              size_t ws_size, hipStream_t stream) {
  const float* x   = (const float*)d_in[0];  // N x 128
  const int*   ei  = (const int*)d_in[1];    // 2 x E (row 0 = src, row 1 = dst)
  const float* W   = (const float*)d_in[2];  // 128 x 128
  const float* att = (const float*)d_in[3];  // 1 x 4 x 64
  float* out = (float*)d_out;                // N x 128

  const int N = in_sizes[0] / GAT_D;  // 50000
  const int E = in_sizes[1] / 2;      // 1600000
  const int NH = N * GAT_H;

  const int* srcI = ei;
  const int* dstI = ei + E;

  // workspace layout (28.8 MB total)
  float*    xh     = (float*)d_ws;                        // N*128 f32
  float*    s_d    = xh + (size_t)N * GAT_D;              // N*4 f32
  float*    s_s    = s_d + (size_t)NH;                    // N*4 f32
  unsigned* amax_u = (unsigned*)(s_s + (size_t)NH);       // N*4 u32
  float*    denom  = (float*)(amax_u + (size_t)NH);       // N*4 f32

  const int n_out = N * GAT_D;

  // 0: init
  gat_init<<<(n_out + 255) / 256, 256, 0, stream>>>(out, amax_u, denom, n_out,
                                                    NH);
  // 1: WMMA GEMM (N/16 = 3125 full blocks)
  gat_gemm_wmma<<<N / 16, 256, 0, stream>>>(x, W, xh);
  // 2: node scores
  gat_scores<<<(NH + 255) / 256, 256, 0, stream>>>(xh, att, s_d, s_s, NH);
  // 3: segment max
  gat_edge_max<<<(E + 255) / 256, 256, 0, stream>>>(srcI, dstI, s_d, s_s,
                                                    amax_u, E);
  // 4: denominator
  gat_edge_sum<<<(E + 255) / 256, 256, 0, stream>>>(srcI, dstI, s_d, s_s,
                                                    amax_u, denom, E);
  // 5: scatter messages (one wave per edge; E/8 = 200000 full blocks)
  gat_scatter<<<E / 8, 256, 0, stream>>>(xh, srcI, dstI, s_d, s_s, amax_u,
                                         denom, out, E);
}